// Attention_84215718740239
// MI455X (gfx1250) — compile-verified
//
#include <hip/hip_runtime.h>
#include <math.h>
#include <stdint.h>

// ---------------------------------------------------------------------------
// Problem constants (from reference): B=16, N=512, C=512, H=8, hd=64
// ---------------------------------------------------------------------------
#define BATCH   16
#define SEQ     512
#define CH      512
#define HEADS   8
#define HDIM    64
#define NB      (BATCH * HEADS)          // 128
#define LENF    (SEQ / 2)                // 256
#define FLATLEN (SEQ * LENF)             // 131072 per batch-head
#define SAMPLEK 12800                    // int(0.1*(256-1))*512

typedef float v2f __attribute__((ext_vector_type(2)));
typedef float v8f __attribute__((ext_vector_type(8)));

// ---------------------------------------------------------------------------
// f32 WMMA: D = A(16x4) * B(4x16) + C(16x16), full f32 precision.
// Chosen over f16/bf16 WMMA because the Gumbel top-k mask is discontinuous in
// the logits; low-precision logits would flip the selected indices and change
// the output macroscopically.
// ---------------------------------------------------------------------------
static __device__ __forceinline__ v8f wmma_f32_4(v2f a, v2f b, v8f c) {
#if defined(__gfx1250__) && __has_builtin(__builtin_amdgcn_wmma_f32_16x16x4_f32)
  return __builtin_amdgcn_wmma_f32_16x16x4_f32(false, a, false, b, (short)0, c,
                                               false, false);
#else
  asm volatile("v_wmma_f32_16x16x4_f32 %0, %1, %2, %0"
               : "+v"(c)
               : "v"(a), "v"(b));
  return c;
#endif
}

// ---------------------------------------------------------------------------
// Generic batched WMMA GEMM, 64x64 block tile.
//   BT=true : C = alpha * A[MxK] * B[NxK]^T + bias   (row-major A, B)
//   BT=false: C = alpha * A[MxK] * B[KxN]   + bias
// Block = 256 threads = 8 waves arranged 4x2; each wave owns a 16x32 slab
// (two 16x16 f32 accumulators sharing the A fragment -> two independent WMMA
// chains). A and B K-chunks staged in LDS:
//   * float4 (b128) global loads, fully coalesced
//   * both tiles stored row-major-per-fragment (As[m][k], Bs[n][k]) with row
//     stride 36 dwords: every fragment is one aligned 8-byte LDS load and the
//     32 lanes x 2 dwords of a fragment read touch all 64 banks exactly once.
// M, N multiples of 64 and K multiple of 32 (true for every use here), so
// EXEC is all-ones around WMMA with no guards.
// ---------------------------------------------------------------------------
template <bool BT>
__global__ __launch_bounds__(256)
void wmma_gemm(const float* __restrict__ A, const float* __restrict__ Bm,
               const float* __restrict__ bias, float* __restrict__ C,
               int K, int lda, int ldb, int ldc, float alpha,
               long long aBatch, long long bBatch, int cDivH,
               long long cOuter, long long cInner) {
  const int z = blockIdx.z;
  A  += (long long)z * aBatch;
  Bm += (long long)z * bBatch;
  C  += (long long)(z / cDivH) * cOuter + (long long)(z % cDivH) * cInner;

  const int t     = threadIdx.x;
  const int wave  = t >> 5;                // 0..7
  const int lane  = t & 31;
  const int l16   = lane & 15;
  const int khalf = (lane >> 4) * 2;       // 0 or 2
  const int wr    = wave >> 1;             // 0..3  (wave row)
  const int wc    = wave & 1;              // 0..1  (wave col)

  const int m0  = blockIdx.y * 64;
  const int n0g = blockIdx.x * 64;

  __shared__ __align__(16) float As[64][36];   // As[m][k], padded stride 36
  __shared__ __align__(16) float Bs[64][36];   // Bs[n][k], padded stride 36

  const int aRow = wr * 16 + l16;
  const int bCol = wc * 32 + l16;

  v8f acc0 = {0.f, 0.f, 0.f, 0.f, 0.f, 0.f, 0.f, 0.f};
  v8f acc1 = {0.f, 0.f, 0.f, 0.f, 0.f, 0.f, 0.f, 0.f};

  for (int kc = 0; kc < K; kc += 32) {
    // ---- stage A: 64x32, two float4 per thread, coalesced ----
#pragma unroll
    for (int e = 0; e < 2; ++e) {
      const int idx = t + e * 256;           // quad index 0..511
      const int r = idx >> 3;                // 0..63
      const int c = (idx & 7) * 4;           // 0..28
      const float4 val =
          *(const float4*)&A[(long long)(m0 + r) * lda + (kc + c)];
      *(float4*)&As[r][c] = val;
    }
    // ---- stage B: 32x64 worth of data into Bs[n][k] ----
    if (BT) {                                // B row-major [N][K]: same pattern as A
#pragma unroll
      for (int e = 0; e < 2; ++e) {
        const int idx = t + e * 256;
        const int n = idx >> 3;              // 0..63
        const int c = (idx & 7) * 4;         // 0..28
        const float4 val =
            *(const float4*)&Bm[(long long)(n0g + n) * ldb + (kc + c)];
        *(float4*)&Bs[n][c] = val;
      }
    } else {                                 // B [K][N]: read along N, transpose in LDS
#pragma unroll
      for (int e = 0; e < 2; ++e) {
        const int idx = t + e * 256;
        const int kq = idx >> 4;             // 0..31
        const int n = (idx & 15) * 4;        // 0..60
        const float4 val =
            *(const float4*)&Bm[(long long)(kc + kq) * ldb + (n0g + n)];
        Bs[n + 0][kq] = val.x;
        Bs[n + 1][kq] = val.y;
        Bs[n + 2][kq] = val.z;
        Bs[n + 3][kq] = val.w;
      }
    }
    if (kc + 32 < K) {                       // prefetch next chunks
      __builtin_prefetch(&A[(long long)(m0 + (t >> 2)) * lda + kc + 32], 0, 0);
      if (BT)
        __builtin_prefetch(&Bm[(long long)(n0g + (t >> 2)) * ldb + kc + 32], 0, 0);
      else
        __builtin_prefetch(&Bm[(long long)(kc + 32 + (t >> 3)) * ldb + n0g], 0, 0);
    }
    __syncthreads();

    // ---- 8 K-steps of 4; one aligned 8B LDS load per fragment ----
#pragma unroll
    for (int kk = 0; kk < 32; kk += 4) {
      const v2f a  = *(const v2f*)&As[aRow][kk + khalf];
      const v2f b0 = *(const v2f*)&Bs[bCol][kk + khalf];
      const v2f b1 = *(const v2f*)&Bs[bCol + 16][kk + khalf];
      acc0 = wmma_f32_4(a, b0, acc0);
      acc1 = wmma_f32_4(a, b1, acc1);
    }
    __syncthreads();
  }

  // ---- epilogue ----
  const int cn = n0g + wc * 32 + l16;
  const float bv0 = bias ? bias[cn] : 0.0f;
  const float bv1 = bias ? bias[cn + 16] : 0.0f;
  const int mBase = m0 + wr * 16 + (lane >> 4) * 8;  // C: VGPR r -> M=r (+8 hi)
#pragma unroll
  for (int r = 0; r < 8; ++r) {
    C[(long long)(mBase + r) * ldc + cn]      = alpha * acc0[r] + bv0;
    C[(long long)(mBase + r) * ldc + cn + 16] = alpha * acc1[r] + bv1;
  }
}

// ---------------------------------------------------------------------------
// Scatter fused-QKV [B*N, 3*C] -> q,k,v each [B,H,N,hd]
// ---------------------------------------------------------------------------
__global__ __launch_bounds__(256)
void reorder_qkv(const float* __restrict__ qkv, float* __restrict__ q,
                 float* __restrict__ k, float* __restrict__ v) {
  const int col = blockIdx.x * 256 + threadIdx.x;     // 0..1535
  const long long row = blockIdx.y;                   // 0..8191
  const float val = qkv[row * (3 * CH) + col];
  const int b = (int)(row >> 9), n = (int)(row & (SEQ - 1));
  const int which = col >> 9;                         // /512
  const int rest = col & (CH - 1);
  const int h = rest >> 6, d = rest & (HDIM - 1);
  const long long dst = ((((long long)b * HEADS + h) * SEQ) + n) * HDIM + d;
  float* o = (which == 0) ? q : (which == 1) ? k : v;
  o[dst] = val;
}

// ---------------------------------------------------------------------------
// Block reduction helper (256 threads)
// ---------------------------------------------------------------------------
static __device__ __forceinline__ float block_reduce(float val, int op,
                                                     float* red) {
  const int t = threadIdx.x;
  red[t] = val;
  __syncthreads();
#pragma unroll
  for (int s = 128; s > 0; s >>= 1) {
    if (t < s) {
      const float a = red[t], b = red[t + s];
      red[t] = (op == 0) ? fmaxf(a, b) : (op == 1) ? (a + b) : fminf(a, b);
    }
    __syncthreads();
  }
  const float r = red[0];
  __syncthreads();
  return r;
}

// ---------------------------------------------------------------------------
// Per-row: softmax of logits, then att_final[row, 0..255] per reference:
//   aqm   = max(cross-block softmax) + 1e-6
//   p     = same-block softmax + relu(-min) + 1e-6, zero diagonal
//   att_f = p / sum(p) * aqm
// ---------------------------------------------------------------------------
__global__ __launch_bounds__(256)
void softmax_attfinal(const float* __restrict__ S, float* __restrict__ AF) {
  __shared__ float red[256];
  const long long row = blockIdx.x;
  const int t = threadIdx.x;
  const float* srow = S + row * SEQ;
  const float s0 = srow[t], s1 = srow[t + LENF];

  const float m = block_reduce(fmaxf(s0, s1), 0, red);
  const float e0 = __expf(s0 - m), e1 = __expf(s1 - m);
  const float sum = block_reduce(e0 + e1, 1, red);
  const float inv = 1.0f / sum;

  const int i = (int)(row & (SEQ - 1));
  const int side = i >> 8;            // 0: pre rows, 1: lat rows
  const int il = i & (LENF - 1);

  const float crossE = side ? e0 : e1;                // opposite block
  const float aqm = block_reduce(crossE * inv, 0, red) + 1e-6f;

  float p = (side ? e1 : e0) * inv;                   // same-side block
  const float minv = block_reduce(p, 2, red);
  p = p + fmaxf(-minv, 0.0f) + 1e-6f;
  if (t == il) p = 0.0f;                              // zero diagonal
  const float sp = block_reduce(p, 1, red);

  AF[row * LENF + t] = p / sp * aqm;
}

// ---------------------------------------------------------------------------
// In-place: att_final -> sortable u32 key of (log(att_final) + Gumbel noise).
// Counter-based splitmix64 RNG -> deterministic across graph replays.
// ---------------------------------------------------------------------------
__global__ __launch_bounds__(256)
void gumbel_keys(float* __restrict__ AF) {
  const long long idx = (long long)blockIdx.x * 256 + threadIdx.x;
  const float f = AF[idx];

  unsigned long long x = (unsigned long long)idx + 0x9E3779B97F4A7C15ULL;
  x ^= x >> 30; x *= 0xBF58476D1CE4E5B9ULL;
  x ^= x >> 27; x *= 0x94D049BB133111EBULL;
  x ^= x >> 31;
  const float u = (float)((x >> 40) + 1) * (1.0f / 16777218.0f);  // (0,1)
  const float g = -__logf(-__logf(u));
  const float sc = __logf(f) + g;     // f==0 on diagonal -> -inf, never picked

  const unsigned int b = __float_as_uint(sc);
  const unsigned int key = (b & 0x80000000u) ? ~b : (b | 0x80000000u);
  ((unsigned int*)AF)[idx] = key;
}

// ---------------------------------------------------------------------------
// Deterministic MSB radix-select of the SAMPLEK-th largest key per batch-head.
// One 1024-thread block per nb; 32 counting passes with LDS tree reduction.
// ---------------------------------------------------------------------------
__global__ __launch_bounds__(1024)
void select_kth(const unsigned int* __restrict__ keys,
                unsigned int* __restrict__ kth) {
  __shared__ unsigned int cnt[1024];
  __shared__ unsigned int sPrefix;
  __shared__ int sRem;
  const int nb = blockIdx.x;
  const unsigned int* kb = keys + (long long)nb * FLATLEN;
  const int t = threadIdx.x;
  if (t == 0) { sPrefix = 0u; sRem = SAMPLEK; }
  __syncthreads();

  for (int bit = 31; bit >= 0; --bit) {
    const unsigned int cand = sPrefix | (1u << bit);
    const unsigned int target = cand >> bit;
    unsigned int c = 0;
    for (int e = t; e < FLATLEN; e += 1024)
      c += ((kb[e] >> bit) == target) ? 1u : 0u;
    cnt[t] = c;
    __syncthreads();
#pragma unroll
    for (int s = 512; s > 0; s >>= 1) {
      if (t < s) cnt[t] += cnt[t + s];
      __syncthreads();
    }
    if (t == 0) {
      if ((int)cnt[0] >= sRem) sPrefix = cand;
      else                     sRem -= (int)cnt[0];
    }
    __syncthreads();
  }
  if (t == 0) kth[nb] = sPrefix;
}

// ---------------------------------------------------------------------------
// Masked softmax, in place on logits S -> probabilities P.
// Mask: same-side block column is dropped when its key >= kth[nb].
// ---------------------------------------------------------------------------
__global__ __launch_bounds__(256)
void masked_softmax(float* __restrict__ S, const unsigned int* __restrict__ keys,
                    const unsigned int* __restrict__ kth) {
  __shared__ float red[256];
  const long long row = blockIdx.x;
  const int t = threadIdx.x;
  const int nb = (int)(row >> 9);
  float* srow = S + row * SEQ;
  float s0 = srow[t], s1 = srow[t + LENF];

  const bool msk = keys[row * LENF + t] >= kth[nb];
  const int side = ((int)(row & (SEQ - 1))) >> 8;
  if (msk) {
    if (side) s1 = -INFINITY;   // same-side block is cols [256,512)
    else      s0 = -INFINITY;   // same-side block is cols [0,256)
  }

  const float m = block_reduce(fmaxf(s0, s1), 0, red);
  const float e0 = __expf(s0 - m), e1 = __expf(s1 - m);
  const float sum = block_reduce(e0 + e1, 1, red);
  const float inv = 1.0f / sum;
  srow[t] = e0 * inv;
  srow[t + LENF] = e1 * inv;
}

// ---------------------------------------------------------------------------
// Host-side orchestration
// ---------------------------------------------------------------------------
extern "C" void kernel_launch(void* const* d_in, const int* in_sizes, int n_in,
                              void* d_out, int out_size, void* d_ws,
                              size_t ws_size, hipStream_t stream) {
  (void)in_sizes; (void)n_in; (void)out_size; (void)ws_size;

  const float* x      = (const float*)d_in[0];   // [16,512,512]
  const float* w_qkv  = (const float*)d_in[1];   // [1536,512]
  const float* w_proj = (const float*)d_in[2];   // [512,512]
  const float* b_proj = (const float*)d_in[3];   // [512]
  float* out = (float*)d_out;                    // [16,512,512]

  // Workspace layout
  char* ws = (char*)d_ws;
  float* qkvbuf = (float*)ws;                 ws += (size_t)8192 * 1536 * 4;  // 50.3 MB
  float* qb = (float*)ws;                     ws += (size_t)NB * SEQ * HDIM * 4;
  float* kb = (float*)ws;                     ws += (size_t)NB * SEQ * HDIM * 4;
  float* vb = (float*)ws;                     ws += (size_t)NB * SEQ * HDIM * 4;
  float* S  = (float*)ws;                     ws += (size_t)NB * SEQ * SEQ * 4;  // 128 MB
  float* AF = (float*)ws;                     ws += (size_t)NB * SEQ * LENF * 4; // 64 MB (reused as u32 keys)
  unsigned int* kth = (unsigned int*)ws;      ws += 1024;
  float* ctx = (float*)ws;                    ws += (size_t)BATCH * SEQ * CH * 4;

  const long long bhStride = (long long)SEQ * HDIM;      // 32768
  const long long attStride = (long long)SEQ * SEQ;      // 262144

  // 1) QKV: [8192x512] x [1536x512]^T -> [8192x1536]
  wmma_gemm<true><<<dim3(24, 128, 1), 256, 0, stream>>>(
      x, w_qkv, nullptr, qkvbuf, CH, CH, CH, 3 * CH, 1.0f,
      0LL, 0LL, 1, 0LL, 0LL);

  // 2) split into q,k,v [B,H,N,hd]
  reorder_qkv<<<dim3(6, 8192, 1), 256, 0, stream>>>(qkvbuf, qb, kb, vb);

  // 3) logits S = scale * q @ k^T per (b,h): 128 x [512x512x64]
  wmma_gemm<true><<<dim3(8, 8, NB), 256, 0, stream>>>(
      qb, kb, nullptr, S, HDIM, HDIM, HDIM, SEQ, 0.125f,
      bhStride, bhStride, 1, attStride, 0LL);

  // 4) softmax + att_final construction (one block per attention row)
  softmax_attfinal<<<dim3(NB * SEQ, 1, 1), 256, 0, stream>>>(S, AF);

  // 5) Gumbel keys, in place over AF
  gumbel_keys<<<dim3((NB * FLATLEN) / 256, 1, 1), 256, 0, stream>>>(AF);

  // 6) k-th largest key per batch-head (radix select)
  select_kth<<<dim3(NB, 1, 1), 1024, 0, stream>>>((const unsigned int*)AF, kth);

  // 7) masked softmax in place: S -> P
  masked_softmax<<<dim3(NB * SEQ, 1, 1), 256, 0, stream>>>(
      S, (const unsigned int*)AF, kth);

  // 8) ctx = P @ v per (b,h); write merged as [B, N, H*hd]
  wmma_gemm<false><<<dim3(1, 8, NB), 256, 0, stream>>>(
      S, vb, nullptr, ctx, SEQ, SEQ, HDIM, CH, 1.0f,
      attStride, bhStride, HEADS, (long long)SEQ * CH, (long long)HDIM);

  // 9) out = ctx @ w_proj^T + b_proj
  wmma_gemm<true><<<dim3(8, 128, 1), 256, 0, stream>>>(
      ctx, w_proj, b_proj, out, CH, CH, CH, CH, 1.0f,
      0LL, 0LL, 1, 0LL, 0LL);
}